// Graph2Cone_49572512530720
// MI455X (gfx1250) — compile-verified
//
#include <hip/hip_runtime.h>
#include <math.h>

#define N_NODES 50000
#define N_EDGES 800000
#define C_IN    128
#define H_DIM   256
#define G_GRAPHS 64
#define NEG_SLOPE 0.2f
#define BN_EPS  1e-5f
#define PI_F    3.14159265358979f

typedef __attribute__((ext_vector_type(2))) float v2f;
typedef __attribute__((ext_vector_type(8))) float v8f;

// ---------------------------------------------------------------------------
// f32 WMMA GEMM: Out[M,256] = act(In[M,K] @ W[K,256] + bias)
// One wave -> a 16x64 output strip (4 tiles of 16x16) via V_WMMA_F32_16X16X4_F32.
// A fragment loaded once per k-step and reused across the 4 N-tiles (4x less
// A traffic than 1-tile-per-wave); B (<=256KB) stays cache-resident.
// M multiple of 16, K multiple of 4, N fixed = 256 (4 strips).
// act: 0 = none, 1 = tanh
// ---------------------------------------------------------------------------
__global__ void gemm_wmma_kernel(const float* __restrict__ In,
                                 const float* __restrict__ W,
                                 const float* __restrict__ bias,
                                 float* __restrict__ Out,
                                 int M, int K, int act)
{
    const int lane = threadIdx.x & 31;
    const int wid  = threadIdx.x >> 5;
    const long wgid = (long)blockIdx.x * 8 + wid;   // global wave id
    const long tm = wgid >> 2;                      // M tile row
    const int  tg = (int)(wgid & 3);                // N strip: 64 cols
    if (tm * 16 >= (long)M) return;                 // uniform per-wave

    const int rc = lane & 15;                 // A row / B-D column within tile
    const int kq = (lane >> 4) << 1;          // 0 or 2 (K sub-offset)

    const float* __restrict__ a_ptr = In + (tm * 16 + rc) * (long)K + kq;
    const float* __restrict__ b_ptr = W + (long)kq * H_DIM + tg * 64 + rc;

    v8f acc0 = {}, acc1 = {}, acc2 = {}, acc3 = {};
    for (int k = 0; k < K; k += 4) {
        v2f a;
        a.x = a_ptr[k];                       // A[row][k+kq]
        a.y = a_ptr[k + 1];                   // A[row][k+kq+1]
        const float* bp = b_ptr + (long)k * H_DIM;
        v2f b0, b1, b2, b3;
        b0.x = bp[0];          b0.y = bp[H_DIM];
        b1.x = bp[16];         b1.y = bp[H_DIM + 16];
        b2.x = bp[32];         b2.y = bp[H_DIM + 32];
        b3.x = bp[48];         b3.y = bp[H_DIM + 48];
        acc0 = __builtin_amdgcn_wmma_f32_16x16x4_f32(false, a, false, b0, (short)0, acc0, false, false);
        acc1 = __builtin_amdgcn_wmma_f32_16x16x4_f32(false, a, false, b1, (short)0, acc1, false, false);
        acc2 = __builtin_amdgcn_wmma_f32_16x16x4_f32(false, a, false, b2, (short)0, acc2, false, false);
        acc3 = __builtin_amdgcn_wmma_f32_16x16x4_f32(false, a, false, b3, (short)0, acc3, false, false);
    }

    const long mBase = tm * 16 + ((lane >> 4) << 3);   // +0 or +8
    v8f accs[4] = {acc0, acc1, acc2, acc3};
    #pragma unroll
    for (int t = 0; t < 4; ++t) {
        const int  nCol = tg * 64 + t * 16 + rc;
        const float bv  = bias ? bias[nCol] : 0.0f;
        #pragma unroll
        for (int i = 0; i < 8; ++i) {
            float v = accs[t][i] + bv;
            if (act) v = tanhf(v);
            Out[(mBase + i) * (long)H_DIM + nCol] = v;
        }
    }
}

// ---------------------------------------------------------------------------
__global__ void fill_kernel(float* p, float v, long n) {
    long i = (long)blockIdx.x * blockDim.x + threadIdx.x;
    if (i < n) p[i] = v;
}

__global__ void deg_kernel(const int* __restrict__ dst, float* __restrict__ cnt, int E) {
    int e = blockIdx.x * 256 + threadIdx.x;
    if (e < E) atomicAdd(&cnt[dst[e]], 1.0f);
}

// loopW[n][:] += eW[e][:]  for dst[e]==n   (1 block per edge, 256 channels)
__global__ void segsum_kernel(const int* __restrict__ dst,
                              const float* __restrict__ eW,
                              float* __restrict__ loopW) {
    long e = blockIdx.x; int h = threadIdx.x;
    atomicAdd(&loopW[(long)dst[e] * H_DIM + h], eW[e * H_DIM + h]);
}

__global__ void div_cnt_kernel(float* __restrict__ loopW, const float* __restrict__ cnt) {
    long i = blockIdx.x; int h = threadIdx.x;
    loopW[i * H_DIM + h] /= fmaxf(cnt[i], 1.0f);
}

__device__ inline float wave_sum32(float v) {
    #pragma unroll
    for (int off = 16; off > 0; off >>= 1) v += __shfl_down(v, off, 32);
    return v;
}

// logit[e] = sum_h leaky(xl[src][h] + xr[dst][h] + eW[e][h]) * att[h]
__global__ void edge_logit_kernel(const int* __restrict__ src, const int* __restrict__ dst,
                                  const float* __restrict__ xl, const float* __restrict__ xr,
                                  const float* __restrict__ eW, const float* __restrict__ att,
                                  float* __restrict__ logit, int E)
{
    int e = blockIdx.x * 8 + (threadIdx.x >> 5);
    if (e >= E) return;
    int lane = threadIdx.x & 31;
    const float* pl = xl + (long)src[e] * H_DIM;
    const float* pr = xr + (long)dst[e] * H_DIM;
    const float* pe = eW + (long)e * H_DIM;
    float s = 0.f;
    for (int h = lane; h < H_DIM; h += 32) {
        float m = pl[h] + pr[h] + pe[h];
        m = (m > 0.f) ? m : NEG_SLOPE * m;
        s += m * att[h];
    }
    s = wave_sum32(s);
    if (lane == 0) logit[e] = s;
}

// self-loop logit per node: m = leaky(xl[n] + xr[n] + loopW[n])
__global__ void self_logit_kernel(const float* __restrict__ xl, const float* __restrict__ xr,
                                  const float* __restrict__ loopW, const float* __restrict__ att,
                                  float* __restrict__ lself, int n)
{
    int i = blockIdx.x * 8 + (threadIdx.x >> 5);
    if (i >= n) return;
    int lane = threadIdx.x & 31;
    const float* pl = xl + (long)i * H_DIM;
    const float* pr = xr + (long)i * H_DIM;
    const float* pe = loopW + (long)i * H_DIM;
    float s = 0.f;
    for (int h = lane; h < H_DIM; h += 32) {
        float m = pl[h] + pr[h] + pe[h];
        m = (m > 0.f) ? m : NEG_SLOPE * m;
        s += m * att[h];
    }
    s = wave_sum32(s);
    if (lane == 0) lself[i] = s;
}

__device__ inline void atomicMaxFloat(float* addr, float val) {
    if (val >= 0.0f) atomicMax((int*)addr, __float_as_int(val));
    else             atomicMin((unsigned int*)addr, __float_as_uint(val));
}

__global__ void copy_kernel(const float* __restrict__ a, float* __restrict__ b, long n) {
    long i = (long)blockIdx.x * blockDim.x + threadIdx.x;
    if (i < n) b[i] = a[i];
}

__global__ void edge_max_kernel(const int* __restrict__ dst, const float* __restrict__ logit,
                                float* __restrict__ nmax, int E) {
    int e = blockIdx.x * 256 + threadIdx.x;
    if (e < E) atomicMaxFloat(&nmax[dst[e]], logit[e]);
}

__global__ void self_exp_kernel(const float* __restrict__ nmax, float* __restrict__ lself,
                                float* __restrict__ ssum, int n) {
    int i = blockIdx.x * 256 + threadIdx.x;
    if (i >= n) return;
    float v = expf(lself[i] - nmax[i]);
    lself[i] = v;
    ssum[i]  = v;
}

__global__ void edge_exp_kernel(const int* __restrict__ dst, const float* __restrict__ nmax,
                                float* __restrict__ logit, float* __restrict__ ssum, int E) {
    int e = blockIdx.x * 256 + threadIdx.x;
    if (e >= E) return;
    int d = dst[e];
    float v = expf(logit[e] - nmax[d]);
    logit[e] = v;
    atomicAdd(&ssum[d], v);
}

// out[n][h] = alpha_self * xl[n][h] + bias[h]
__global__ void out_init_kernel(const float* __restrict__ xl, const float* __restrict__ lself,
                                const float* __restrict__ ssum, const float* __restrict__ bias,
                                float* __restrict__ out) {
    long i = blockIdx.x; int h = threadIdx.x;
    float a = lself[i] / (ssum[i] + 1e-16f);
    out[i * H_DIM + h] = a * xl[i * H_DIM + h] + bias[h];
}

// out[dst][h] += alpha_e * xl[src][h]
__global__ void scatter_kernel(const int* __restrict__ src, const int* __restrict__ dst,
                               const float* __restrict__ xl, const float* __restrict__ logit,
                               const float* __restrict__ ssum, float* __restrict__ out) {
    long e = blockIdx.x; int h = threadIdx.x;
    int s = src[e], d = dst[e];
    float a = logit[e] / (ssum[d] + 1e-16f);
    atomicAdd(&out[(long)d * H_DIM + h], a * xl[(long)s * H_DIM + h]);
}

// one block per channel: mean + var over N rows
__global__ void bn_stats_kernel(const float* __restrict__ x, float* __restrict__ mu,
                                float* __restrict__ var, int n) {
    int ch = blockIdx.x;
    float s = 0.f, s2 = 0.f;
    for (int i = threadIdx.x; i < n; i += 256) {
        float v = x[(long)i * H_DIM + ch];
        s += v; s2 += v * v;
    }
    __shared__ float sh[256], sh2[256];
    sh[threadIdx.x] = s; sh2[threadIdx.x] = s2;
    __syncthreads();
    for (int o = 128; o > 0; o >>= 1) {
        if (threadIdx.x < o) { sh[threadIdx.x] += sh[threadIdx.x + o];
                               sh2[threadIdx.x] += sh2[threadIdx.x + o]; }
        __syncthreads();
    }
    if (threadIdx.x == 0) {
        float m = sh[0] / n;
        mu[ch] = m;
        var[ch] = sh2[0] / n - m * m;
    }
}

__global__ void bn_tanh_kernel(float* __restrict__ x, const float* __restrict__ mu,
                               const float* __restrict__ var, const float* __restrict__ g,
                               const float* __restrict__ b) {
    long i = blockIdx.x; int h = threadIdx.x;
    long idx = i * H_DIM + h;
    float v = (x[idx] - mu[h]) * rsqrtf(var[h] + BN_EPS) * g[h] + b[h];
    x[idx] = tanhf(v);
}

__global__ void pool_max_kernel(const int* __restrict__ batch, const float* __restrict__ gate,
                                float* __restrict__ gmax) {
    long i = blockIdx.x; int h = threadIdx.x;
    atomicMaxFloat(&gmax[(long)batch[i] * H_DIM + h], gate[i * H_DIM + h]);
}

__global__ void pool_exp_kernel(const int* __restrict__ batch, float* __restrict__ gate,
                                const float* __restrict__ gmax, float* __restrict__ gsum) {
    long i = blockIdx.x; int h = threadIdx.x;
    long gi = (long)batch[i] * H_DIM + h;
    float v = expf(gate[i * H_DIM + h] - gmax[gi]);
    gate[i * H_DIM + h] = v;
    atomicAdd(&gsum[gi], v);
}

__global__ void pool_acc_kernel(const int* __restrict__ batch, const float* __restrict__ gate,
                                const float* __restrict__ gsum, const float* __restrict__ hfeat,
                                float* __restrict__ pooled) {
    long i = blockIdx.x; int h = threadIdx.x;
    long gi = (long)batch[i] * H_DIM + h;
    float w = gate[i * H_DIM + h] / (gsum[gi] + 1e-16f);
    atomicAdd(&pooled[gi], w * hfeat[i * H_DIM + h]);
}

// split y[G,256] -> axis = y[:, :128]*PI ; aperture = (y[:,128:]+1)*PI
__global__ void split_kernel(const float* __restrict__ y, float* __restrict__ out) {
    int g = blockIdx.x; int j = threadIdx.x;
    float v = y[(long)g * H_DIM + j];
    if (j < 128) out[(long)g * 128 + j] = v * PI_F;
    else         out[(long)G_GRAPHS * 128 + (long)g * 128 + (j - 128)] = (v + 1.0f) * PI_F;
}

// ---------------------------------------------------------------------------
static inline void launch_gemm(hipStream_t s, const float* In, const float* W,
                               const float* bias, float* Out, int M, int K, int act) {
    int waves  = M / 4;              // (M/16) tile rows * 4 strips
    int blocks = (waves + 7) / 8;    // 8 waves per block
    gemm_wmma_kernel<<<blocks, 256, 0, s>>>(In, W, bias, Out, M, K, act);
}

static void run_gat_layer(hipStream_t st,
    const float* xin, int Kin, const float* ea,
    const int* src, const int* dst, const float* cnt,
    const float* Wl, const float* bl, const float* Wr, const float* br,
    const float* We, const float* att, const float* bias,
    const float* gam, const float* bet,
    float* xl, float* xr, float* eW, float* loopW,
    float* logit, float* lself, float* nmax, float* ssum,
    float* mu, float* var, float* hout)
{
    const long NH = (long)N_NODES * H_DIM;
    // dense transforms (WMMA)
    launch_gemm(st, xin, Wl, bl, xl, N_NODES, Kin, 0);
    launch_gemm(st, xin, Wr, br, xr, N_NODES, Kin, 0);
    launch_gemm(st, ea,  We, nullptr, eW, N_EDGES, C_IN, 0);
    // loopW = segment_sum(eW, dst) / max(cnt,1)   (linearity of @We)
    fill_kernel<<<(int)((NH + 255) / 256), 256, 0, st>>>(loopW, 0.0f, NH);
    segsum_kernel<<<N_EDGES, 256, 0, st>>>(dst, eW, loopW);
    div_cnt_kernel<<<N_NODES, 256, 0, st>>>(loopW, cnt);
    // attention logits
    edge_logit_kernel<<<(N_EDGES + 7) / 8, 256, 0, st>>>(src, dst, xl, xr, eW, att, logit, N_EDGES);
    self_logit_kernel<<<(N_NODES + 7) / 8, 256, 0, st>>>(xl, xr, loopW, att, lself, N_NODES);
    // segment softmax over dst (self loop included)
    copy_kernel<<<(N_NODES + 255) / 256, 256, 0, st>>>(lself, nmax, N_NODES);
    edge_max_kernel<<<(N_EDGES + 255) / 256, 256, 0, st>>>(dst, logit, nmax, N_EDGES);
    self_exp_kernel<<<(N_NODES + 255) / 256, 256, 0, st>>>(nmax, lself, ssum, N_NODES);
    edge_exp_kernel<<<(N_EDGES + 255) / 256, 256, 0, st>>>(dst, nmax, logit, ssum, N_EDGES);
    // weighted aggregation
    out_init_kernel<<<N_NODES, 256, 0, st>>>(xl, lself, ssum, bias, hout);
    scatter_kernel<<<N_EDGES, 256, 0, st>>>(src, dst, xl, logit, ssum, hout);
    // batchnorm + tanh (in place)
    bn_stats_kernel<<<H_DIM, 256, 0, st>>>(hout, mu, var, N_NODES);
    bn_tanh_kernel<<<N_NODES, 256, 0, st>>>(hout, mu, var, gam, bet);
}

extern "C" void kernel_launch(void* const* d_in, const int* in_sizes, int n_in,
                              void* d_out, int out_size, void* d_ws, size_t ws_size,
                              hipStream_t stream)
{
    (void)in_sizes; (void)n_in; (void)out_size; (void)ws_size;
    const float* x     = (const float*)d_in[0];
    const int*   ei    = (const int*)d_in[1];
    const float* ea    = (const float*)d_in[2];
    const int*   batch = (const int*)d_in[3];
    const float *Wl1 = (const float*)d_in[4],  *bl1  = (const float*)d_in[5];
    const float *Wr1 = (const float*)d_in[6],  *br1  = (const float*)d_in[7];
    const float *We1 = (const float*)d_in[8],  *att1 = (const float*)d_in[9];
    const float *bias1 = (const float*)d_in[10], *g1 = (const float*)d_in[11],
                *be1 = (const float*)d_in[12];
    const float *Wl2 = (const float*)d_in[13], *bl2  = (const float*)d_in[14];
    const float *Wr2 = (const float*)d_in[15], *br2  = (const float*)d_in[16];
    const float *We2 = (const float*)d_in[17], *att2 = (const float*)d_in[18];
    const float *bias2 = (const float*)d_in[19], *g2 = (const float*)d_in[20],
                *be2 = (const float*)d_in[21];
    const float *aW1 = (const float*)d_in[22], *ab1 = (const float*)d_in[23];
    const float *aW2 = (const float*)d_in[24], *ab2 = (const float*)d_in[25];
    const float *lW  = (const float*)d_in[26], *lb  = (const float*)d_in[27];

    const int* src = ei;
    const int* dst = ei + N_EDGES;

    float* ws = (float*)d_ws;
    long off = 0;
    float* eW    = ws + off; off += (long)N_EDGES * H_DIM;
    float* xl    = ws + off; off += (long)N_NODES * H_DIM;
    float* xr    = ws + off; off += (long)N_NODES * H_DIM;
    float* loopW = ws + off; off += (long)N_NODES * H_DIM;
    float* hbuf  = ws + off; off += (long)N_NODES * H_DIM;
    float* hbuf2 = ws + off; off += (long)N_NODES * H_DIM;
    float* logit = ws + off; off += N_EDGES;
    float* lself = ws + off; off += N_NODES;
    float* nmax  = ws + off; off += N_NODES;
    float* ssum  = ws + off; off += N_NODES;
    float* cnt   = ws + off; off += N_NODES;
    float* gmax  = ws + off; off += (long)G_GRAPHS * H_DIM;
    float* gsum  = ws + off; off += (long)G_GRAPHS * H_DIM;
    float* pooled= ws + off; off += (long)G_GRAPHS * H_DIM;
    float* ybuf  = ws + off; off += (long)G_GRAPHS * H_DIM;
    float* mu    = ws + off; off += H_DIM;
    float* var   = ws + off; off += H_DIM;

    const long GH = (long)G_GRAPHS * H_DIM;

    // in-degree (shared by both layers)
    fill_kernel<<<(N_NODES + 255) / 256, 256, 0, stream>>>(cnt, 0.0f, N_NODES);
    deg_kernel<<<(N_EDGES + 255) / 256, 256, 0, stream>>>(dst, cnt, N_EDGES);

    // layer 1: C=128 -> H=256, then BN+tanh into hbuf
    run_gat_layer(stream, x, C_IN, ea, src, dst, cnt,
                  Wl1, bl1, Wr1, br1, We1, att1, bias1, g1, be1,
                  xl, xr, eW, loopW, logit, lself, nmax, ssum, mu, var, hbuf);

    // layer 2: H=256 -> H=256, then BN+tanh into hbuf2
    run_gat_layer(stream, hbuf, H_DIM, ea, src, dst, cnt,
                  Wl2, bl2, Wr2, br2, We2, att2, bias2, g2, be2,
                  xl, xr, eW, loopW, logit, lself, nmax, ssum, mu, var, hbuf2);

    // attentional aggregation: gate = tanh(h@aW1+ab1)@aW2+ab2  (reuse xl, xr)
    launch_gemm(stream, hbuf2, aW1, ab1, xl, N_NODES, H_DIM, 1);   // tanh
    launch_gemm(stream, xl,    aW2, ab2, xr, N_NODES, H_DIM, 0);   // gate in xr
    fill_kernel<<<(int)((GH + 255) / 256), 256, 0, stream>>>(gmax, -3.0e38f, GH);
    fill_kernel<<<(int)((GH + 255) / 256), 256, 0, stream>>>(gsum, 0.0f, GH);
    fill_kernel<<<(int)((GH + 255) / 256), 256, 0, stream>>>(pooled, 0.0f, GH);
    pool_max_kernel<<<N_NODES, 256, 0, stream>>>(batch, xr, gmax);
    pool_exp_kernel<<<N_NODES, 256, 0, stream>>>(batch, xr, gmax, gsum);
    pool_acc_kernel<<<N_NODES, 256, 0, stream>>>(batch, xr, gsum, hbuf2, pooled);

    // y = tanh(pooled @ lW + lb), then split/scale into d_out
    launch_gemm(stream, pooled, lW, lb, ybuf, G_GRAPHS, H_DIM, 1);
    split_kernel<<<G_GRAPHS, 256, 0, stream>>>(ybuf, (float*)d_out);
}